// Criterion_28278064676991
// MI455X (gfx1250) — compile-verified
//
#include <hip/hip_runtime.h>

typedef __attribute__((ext_vector_type(2))) float v2f;
typedef __attribute__((ext_vector_type(8))) float v8f;

#define N_      4096
#define D_      512
#define P_      8
#define Q_      32
#define NSLOT   (P_ + Q_)      /* 40 neighbor slots per anchor */
#define MARGIN_ 1.0f
#define L2W_    0.005f
#define EPS_    1e-6f

// ---------------------------------------------------------------------------
// Kernel 1: per-row stats. One wave (32 lanes) per row.
//   s1[r] = sum_d batch[r][d], s2[r] = sum_d batch[r][d]^2, nrm[r] = sqrt(s2)
// ---------------------------------------------------------------------------
__global__ void __launch_bounds__(256)
row_stats_kernel(const float* __restrict__ batch,
                 float* __restrict__ s1,
                 float* __restrict__ s2,
                 float* __restrict__ nrm) {
  const int wave = threadIdx.x >> 5;            // 0..7
  const int lane = threadIdx.x & 31;
  const int row  = blockIdx.x * 8 + wave;
  if (row >= N_) return;
  const float* r = batch + (size_t)row * D_;
  float a = 0.f, b = 0.f;
#pragma unroll
  for (int j = 0; j < D_ / 32; ++j) {
    float x = r[lane + 32 * j];
    a += x;
    b += x * x;
  }
#pragma unroll
  for (int off = 16; off >= 1; off >>= 1) {
    a += __shfl_xor(a, off, 32);
    b += __shfl_xor(b, off, 32);
  }
  if (lane == 0) {
    s1[row]  = a;
    s2[row]  = b;
    nrm[row] = sqrtf(b);
  }
}

// ---------------------------------------------------------------------------
// Kernel 2: main. One workgroup (4 waves) per tile of 16 anchors.
// For neighbor slot k, a wave computes a 16x16 WMMA (f32 16x16x4, K-chained
// over D=512) whose diagonal holds dot(anchor n, k-th neighbor of anchor n).
// Distance via norm expansion:
//   d^2 = s2[a] + s2[b] - 2*dot + 2*EPS*(s1[a]-s1[b]) + D*EPS^2
// Then 16 lanes do the per-anchor logsumexp / relu.
// ---------------------------------------------------------------------------
__global__ void __launch_bounds__(128)
triplet_wmma_kernel(const float* __restrict__ batch,
                    const int*   __restrict__ anchors,
                    const int*   __restrict__ pos_idx,
                    const int*   __restrict__ neg_idx,
                    const float* __restrict__ s1,
                    const float* __restrict__ s2,
                    float* __restrict__ perAnchor) {
  __shared__ float distBuf[NSLOT][16];

  const int tile  = blockIdx.x;                 // 0..255
  const int tbase = tile * 16;
  const int tid   = threadIdx.x;
  const int wave  = tid >> 5;                   // 0..3
  const int lane  = tid & 31;
  const int m     = lane & 15;                  // matrix row (A) / col (B)
  const int h     = lane >> 4;                  // K-half selector

  // A fragment source: anchor row for this lane's M index.
  const int arow = anchors[tbase + m];
  const float* aP = batch + (size_t)arow * D_ + 2 * h;

  // Each wave owns slots k = wave, wave+4, ... (10 slots each).
  for (int k = wave; k < NSLOT; k += 4) {
    const int nb = (k < P_) ? pos_idx[(tbase + m) * P_ + k]
                            : neg_idx[(tbase + m) * Q_ + (k - P_)];
    const float* bP = batch + (size_t)nb * D_ + 2 * h;

    v8f c = {};
#pragma unroll 8
    for (int kc = 0; kc < D_ / 4; ++kc) {
      // 32-bit A 16x4 layout: lanes 0-15 hold K={0,1}, lanes 16-31 K={2,3}
      // 32-bit B 4x16 layout is the mirror image -> identical lane addressing.
      v2f av = *(const v2f*)(aP + kc * 4);
      v2f bv = *(const v2f*)(bP + kc * 4);
      c = __builtin_amdgcn_wmma_f32_16x16x4_f32(false, av, false, bv,
                                                (short)0, c, false, false);
    }

    // Diagonal extraction from C layout:
    //  lanes 0-15:  N=lane,    M=v     -> diag at lanes 0..7,  c[lane]
    //  lanes 16-31: N=lane-16, M=v+8   -> diag at lanes 24..31, c[lane-24]
    float dot = 0.f;
#pragma unroll
    for (int v = 0; v < 8; ++v) {
      if (lane == v)      dot = c[v];
      if (lane == v + 24) dot = c[v];
    }
    if ((lane < 8) || (lane >= 24)) {
      const int n  = (lane < 8) ? lane : (lane - 16);
      const int ar = anchors[tbase + n];
      const int nbi = (k < P_) ? pos_idx[(tbase + n) * P_ + k]
                               : neg_idx[(tbase + n) * Q_ + (k - P_)];
      float d2 = s2[ar] + s2[nbi] - 2.0f * dot
               + 2.0f * EPS_ * (s1[ar] - s1[nbi])
               + (float)D_ * EPS_ * EPS_;
      d2 = fmaxf(d2, 0.0f);
      distBuf[k][n] = sqrtf(d2);
    }
  }
  __syncthreads();

  if (tid < 16) {
    // pos_term = logsumexp(d_pos)
    float mx = -1e30f;
    for (int k = 0; k < P_; ++k) mx = fmaxf(mx, distBuf[k][tid]);
    float s = 0.f;
    for (int k = 0; k < P_; ++k) s += expf(distBuf[k][tid] - mx);
    const float pos_term = mx + logf(s);
    // neg_term = logsumexp(MARGIN - d_neg)
    float mx2 = -1e30f;
    for (int k = P_; k < NSLOT; ++k) mx2 = fmaxf(mx2, MARGIN_ - distBuf[k][tid]);
    float s2v = 0.f;
    for (int k = P_; k < NSLOT; ++k) s2v += expf(MARGIN_ - distBuf[k][tid] - mx2);
    const float neg_term = mx2 + logf(s2v);
    perAnchor[tbase + tid] = fmaxf(pos_term + neg_term, 0.0f);
  }
}

// ---------------------------------------------------------------------------
// Kernel 3: deterministic final reduction (fixed-order strided + LDS tree).
// ---------------------------------------------------------------------------
__global__ void __launch_bounds__(256)
finalize_kernel(const float* __restrict__ perAnchor,
                const float* __restrict__ nrm,
                float* __restrict__ out) {
  __shared__ float sa[256];
  __shared__ float sb[256];
  const int t = threadIdx.x;
  float a = 0.f, b = 0.f;
  for (int i = t; i < N_; i += 256) { a += perAnchor[i]; b += nrm[i]; }
  sa[t] = a; sb[t] = b;
  __syncthreads();
  for (int off = 128; off >= 1; off >>= 1) {
    if (t < off) { sa[t] += sa[t + off]; sb[t] += sb[t + off]; }
    __syncthreads();
  }
  if (t == 0) out[0] = sa[0] / (float)N_ + L2W_ * (sb[0] / (float)N_);
}

// ---------------------------------------------------------------------------
extern "C" void kernel_launch(void* const* d_in, const int* in_sizes, int n_in,
                              void* d_out, int out_size, void* d_ws, size_t ws_size,
                              hipStream_t stream) {
  (void)in_sizes; (void)n_in; (void)out_size; (void)ws_size;

  const float* batch   = (const float*)d_in[0];   // [N, D] fp32
  const int*   anchors = (const int*)d_in[1];     // [N] int32
  const int*   pos_idx = (const int*)d_in[2];     // [N, P] int32
  const int*   neg_idx = (const int*)d_in[3];     // [N, Q] int32
  float*       out     = (float*)d_out;           // scalar

  float* ws        = (float*)d_ws;
  float* s1        = ws;                          // [N]
  float* s2        = ws + N_;                     // [N]
  float* nrm       = ws + 2 * N_;                 // [N]
  float* perAnchor = ws + 3 * N_;                 // [N]

  row_stats_kernel<<<N_ / 8, 256, 0, stream>>>(batch, s1, s2, nrm);
  triplet_wmma_kernel<<<N_ / 16, 128, 0, stream>>>(batch, anchors, pos_idx,
                                                   neg_idx, s1, s2, perAnchor);
  finalize_kernel<<<1, 256, 0, stream>>>(perAnchor, nrm, out);
}